// SupConLoss_59528246722684
// MI455X (gfx1250) — compile-verified
//
#include <hip/hip_runtime.h>
#include <math.h>

#define NCLS  10
#define NPAD  16      // class-sum matrix zero-padded to 16 rows for WMMA
#define HDIM  768
#define NROWS 8192
#define TEMP  0.07f

typedef __attribute__((ext_vector_type(2))) float v2f;
typedef __attribute__((ext_vector_type(8))) float v8f;

// ---------------- kernel 0: zero the workspace (padded sums + counts) ------
__global__ void supcon_zero_ws(float* ws, int n) {
    int i = blockIdx.x * blockDim.x + threadIdx.x;
    if (i < n) ws[i] = 0.0f;
}

// ---------------- kernel 1: normalize rows, accumulate per-class sums ------
// One wave32 per row: each lane owns columns {lane, lane+32, ...} (24 elems),
// giving 128B-coalesced global loads and conflict-free LDS atomics.
__global__ __launch_bounds__(256) void supcon_accum(
        const float* __restrict__ x,     // [NROWS, HDIM]
        const int*   __restrict__ lab,   // [NROWS]
        float*       __restrict__ gsum,  // [NPAD * HDIM], rows >= NCLS stay 0
        int*         __restrict__ gcnt)  // [NCLS]
{
    __shared__ float lsum[NCLS * HDIM];  // 30 KB
    __shared__ int   lcnt[NCLS];

    const int tid = threadIdx.x;
    for (int i = tid; i < NCLS * HDIM; i += blockDim.x) lsum[i] = 0.0f;
    if (tid < NCLS) lcnt[tid] = 0;
    __syncthreads();

    const int lane   = tid & 31;
    const int wave   = tid >> 5;
    const int wavesPerBlock = blockDim.x >> 5;
    const int waveId = blockIdx.x * wavesPerBlock + wave;
    const int nWaves = gridDim.x * wavesPerBlock;

    for (int row = waveId; row < NROWS; row += nWaves) {
        const float* xr = x + (size_t)row * HDIM;
        float v[24];
        float ss = 0.0f;
        #pragma unroll
        for (int j = 0; j < 24; ++j) {
            v[j] = xr[lane + 32 * j];
            ss += v[j] * v[j];
        }
        // wave32 all-reduce of the squared norm
        #pragma unroll
        for (int m = 16; m >= 1; m >>= 1)
            ss += __shfl_xor(ss, m, 32);
        const float inv = 1.0f / fmaxf(sqrtf(ss), 1e-12f);

        const int c = lab[row];
        float* dst = &lsum[c * HDIM];
        #pragma unroll
        for (int j = 0; j < 24; ++j)
            atomicAdd(&dst[lane + 32 * j], v[j] * inv);
        if (lane == 0) atomicAdd(&lcnt[c], 1);
    }
    __syncthreads();

    for (int i = tid; i < NCLS * HDIM; i += blockDim.x)
        atomicAdd(&gsum[i], lsum[i]);
    if (tid < NCLS) atomicAdd(&gcnt[tid], lcnt[tid]);
}

// ---------------- kernel 2: 16x16 Gram of class sums via f32 WMMA ----------
// S is [NPAD, HDIM] with rows >= NCLS zeroed in the workspace, so every load
// is an unconditional global_load_b64 (no EXEC predication). G = S * S^T via
// V_WMMA_F32_16X16X4_F32, 192 K-steps split across two accumulators to break
// the WMMA->WMMA RAW chain on C. For the symmetric Gram the A (16x4) and
// B (4x16) lane layouts hold identical data, so one register pair feeds both
// operands. sum(G) = ||sum_c S_c||^2 = sum(sim)*T ; trace(G) = pos_sum*T.
__global__ __launch_bounds__(32) void supcon_finalize(
        const float* __restrict__ gsum,  // [NPAD * HDIM]
        const int*   __restrict__ gcnt,  // [NCLS]
        float*       __restrict__ out)
{
    const int lane = threadIdx.x;
    const int m    = lane & 15;       // matrix row this lane carries
    const int kh   = lane >> 4;       // 0 -> K offsets {0,1}, 1 -> {2,3}
    const float* srow = gsum + m * HDIM + 2 * kh;   // 8-byte aligned

    v8f acc0 = {};
    v8f acc1 = {};
    #pragma unroll 4
    for (int kk = 0; kk < HDIM; kk += 8) {
        v2f a0 = *(const v2f*)(srow + kk);
        v2f a1 = *(const v2f*)(srow + kk + 4);
        acc0 = __builtin_amdgcn_wmma_f32_16x16x4_f32(
            false, a0, false, a0, (short)0, acc0, false, false);
        acc1 = __builtin_amdgcn_wmma_f32_16x16x4_f32(
            false, a1, false, a1, (short)0, acc1, false, false);
    }

    // C/D layout: lane holds column n = lane%16; VGPR v holds row v + 8*(lane>=16).
    float tot = 0.0f;
    float dia = 0.0f;
    #pragma unroll
    for (int i = 0; i < 8; ++i) {
        float g = acc0[i] + acc1[i];
        tot += g;
        if ((kh == 0 && m == i) || (kh == 1 && m == i + 8)) dia += g;
    }

    #pragma unroll
    for (int s = 16; s >= 1; s >>= 1) {
        tot += __shfl_xor(tot, s, 32);
        dia += __shfl_xor(dia, s, 32);
    }

    if (lane == 0) {
        float n_pos = 0.0f;
        for (int c = 0; c < NCLS; ++c) {
            float cnt = (float)gcnt[c];
            n_pos += cnt * cnt;
        }
        const float Nf    = (float)NROWS;
        const float n_neg = Nf * Nf - n_pos;
        const float pos_mean = (dia / TEMP) / n_pos;
        const float neg_mean = ((tot - dia) / TEMP) / n_neg;
        const float d = neg_mean - pos_mean;
        // logaddexp(0, d), numerically stable
        out[0] = fmaxf(d, 0.0f) + log1pf(expf(-fabsf(d)));
    }
}

// ---------------------------------------------------------------------------
extern "C" void kernel_launch(void* const* d_in, const int* in_sizes, int n_in,
                              void* d_out, int out_size, void* d_ws, size_t ws_size,
                              hipStream_t stream) {
    const float* x   = (const float*)d_in[0];   // features [16,512,768] f32
    const int*   lab = (const int*)d_in[1];     // labels   [16,512]
    float* out  = (float*)d_out;                // scalar loss, f32
    float* gsum = (float*)d_ws;                 // [NPAD*HDIM], rows>=NCLS stay 0
    int*   gcnt = (int*)(gsum + NPAD * HDIM);   // [NCLS]

    const int ws_elems = NPAD * HDIM + NCLS;
    supcon_zero_ws<<<(ws_elems + 255) / 256, 256, 0, stream>>>(gsum, ws_elems);
    supcon_accum<<<128, 256, 0, stream>>>(x, lab, gsum, gcnt);
    supcon_finalize<<<1, 32, 0, stream>>>(gsum, gcnt, out);
}